// KernelAttnCoef_17351667876294
// MI455X (gfx1250) — compile-verified
//
#include <hip/hip_runtime.h>
#include <hip/hip_bf16.h>

// CDNA5 / gfx1250, wave32. Linear-attention coefficient kernel.
// out[b,k,t,h*8+j] = (sum_i q[b,k,h*16+i] * kv[h,b,t,i,j]) / qk_sum[h,b,k]
// kv[h,b,t,i,j]   = sum_l key[b,l,h*16+i] * value[b,(l*256+t)*64 + h*8+j]
// qk_sum[h,b,k]   = sum_i q[b,k,h*16+i] * (sum_l key[b,l,h*16+i]), 0 -> 1e-5
//
// Bandwidth-bound (~256 MiB traffic, ~2 GFLOP): V is streamed exactly once via
// the Tensor Data Mover (double-buffered 2D tile DMA into LDS), compute uses
// v_wmma_f32_16x16x4_f32, output written coalesced as b128 stores.

typedef float v2f __attribute__((ext_vector_type(2)));
typedef float v8f __attribute__((ext_vector_type(8)));
typedef unsigned int v4u __attribute__((ext_vector_type(4)));
typedef int v4i __attribute__((ext_vector_type(4)));
typedef int v8i __attribute__((ext_vector_type(8)));

namespace {
constexpr int Bn   = 8;    // batch
constexpr int N    = 256;  // sequence
constexpr int D    = 16;   // qk dim per head
constexpr int DV   = 8;    // v dim per head
constexpr int DIMV = 64;   // total v dim
constexpr int F    = 128;  // total qk dim (H*D)
constexpr int TT   = 8;    // t-tile per block
constexpr int NP   = TT / 2;  // t-pairs (2 t's packed into 16 WMMA columns)
constexpr int LCH  = 16;   // l values staged per LDS chunk
constexpr int NCH  = N / LCH;

// dynamic LDS layout (floats)
constexpr int OFF_V0  = 0;      // [LCH][TT][DIMV] = 8192 (TDM buffer 0)
constexpr int OFF_V1  = 8192;   // TDM buffer 1
constexpr int OFF_KV  = 16384;  // [H][TT][D][DV]  = 8192
constexpr int OFF_OUT = 24576;  // [16][TT][DIMV]  = 8192
constexpr int OFF_KS  = 32768;  // ksum[F]         = 128
constexpr int OFF_SC  = 32896;  // scale[H][16]    = 128
constexpr int LDS_FLOATS = 33024;
}

#define WMMA_F32X4(a, b, c) \
  __builtin_amdgcn_wmma_f32_16x16x4_f32(false, (a), false, (b), (short)0, (c), false, false)

#if __has_builtin(__builtin_amdgcn_tensor_load_to_lds) && \
    __has_builtin(__builtin_amdgcn_s_wait_tensorcnt)
#define USE_TDM 1
#else
#define USE_TDM 0
#endif

#if USE_TDM
// DMA one V tile: 2D tile, tile_dim0 = TT*DIMV = 512 f32 (contiguous (t,f) run),
// tile_dim1 = LCH = 16 rows (l), row stride = N*DIMV = 16384 f32. Lands as
// 16x512 contiguous f32 in LDS == lds_v[lc][tt][f].
__device__ __forceinline__ void tdm_load_v_tile(const float* gsrc,
                                                unsigned lds_byte_off) {
  const unsigned long long ga = (unsigned long long)(size_t)gsrc;
  v4u g0;
  g0[0] = 1u;                                   // count=1, user descriptor
  g0[1] = lds_byte_off;                         // lds_addr (bytes)
  g0[2] = (unsigned)(ga & 0xFFFFFFFFu);         // global_addr[31:0]
  g0[3] = (unsigned)((ga >> 32) & 0x01FFFFFFu)  // global_addr[56:32]
          | 0x80000000u;                        // type=2 (bits 127:126 = 0b10)
  v8i g1;
  g1[0] = 0x00020000;                           // data_size=2 (4 bytes)
  g1[1] = (int)(((unsigned)(N * DIMV) & 0xFFFFu) << 16);      // tensor_dim0 lo16
  g1[2] = (int)(((unsigned)(N * DIMV) >> 16) |
                (((unsigned)N & 0xFFFFu) << 16));             // dim0 hi | dim1 lo
  g1[3] = (int)((((unsigned)N >> 16)) |
                (((unsigned)(TT * DIMV)) << 16));             // dim1 hi | tile_dim0
  g1[4] = LCH;                                  // tile_dim1=16, tile_dim2=0
  g1[5] = N * DIMV;                             // tensor_dim0_stride lo32
  g1[6] = 0;                                    // stride0 hi16 | stride1 lo16
  g1[7] = 0;                                    // stride1 hi32
  const v4i z4 = {0, 0, 0, 0};
#if __clang_major__ >= 23
  const v8i z8 = {0, 0, 0, 0, 0, 0, 0, 0};
  __builtin_amdgcn_tensor_load_to_lds(g0, g1, z4, z4, z8, 0);
#else
  __builtin_amdgcn_tensor_load_to_lds(g0, g1, z4, z4, 0);
#endif
}
#endif  // USE_TDM

extern "C" __global__ __launch_bounds__(256)
void KernelAttnCoef_17351667876294_kernel(const float* __restrict__ q,
                                          const float* __restrict__ k,
                                          const float* __restrict__ v,
                                          float* __restrict__ out) {
  extern __shared__ float smem[];
  float* lds_kv  = smem + OFF_KV;
  float* lds_out = smem + OFF_OUT;
  float* ksum    = smem + OFF_KS;
  float* scale   = smem + OFF_SC;

  const int tid  = threadIdx.x;
  const int lane = tid & 31;
  const int h    = tid >> 5;           // wave index == head
  const int b    = blockIdx.x / (N / TT);
  const int t0   = (blockIdx.x % (N / TT)) * TT;

  const float* kb = k + (size_t)b * N * F;
  const float* qb = q + (size_t)b * N * F;
  const float* vb = v + (size_t)b * N * N * DIMV;
  float*       ob = out + (size_t)b * N * N * DIMV;

  // ---- ksum[f] = sum_l key[b,l,f] (threads 0..127, coalesced over f) ----
  if (tid < F) {
    float s = 0.f;
    for (int l = 0; l < N; ++l) s += kb[l * F + tid];
    ksum[tid] = s;
  }

  // Common WMMA lane decomposition
  const int col  = lane & 15;          // N column 0..15
  const int koff = (lane >> 4) * 2;    // K rows handled by this lane half
  const int tsel = col >> 3;           // which t of the pair
  const int j    = col & 7;            // dv index

  // ---- Phase A: kv[t,i,j] accumulation over l, WMMA f32 16x16x4 ----
  v8f acc[NP];
#pragma unroll
  for (int p = 0; p < NP; ++p)
#pragma unroll
    for (int e = 0; e < 8; ++e) acc[p][e] = 0.f;

#if USE_TDM
  const unsigned lds_dyn_base = __builtin_amdgcn_groupstaticsize();
  if (h == 0)  // TDM ignores EXEC; issue once per workgroup, prefill buffer 0
    tdm_load_v_tile(vb + (size_t)t0 * DIMV, lds_dyn_base + OFF_V0 * 4u);
#endif

  for (int c = 0; c < NCH; ++c) {
    const int lc0 = c * LCH;
    float* lds_v = smem + ((c & 1) ? OFF_V1 : OFF_V0);

#if USE_TDM
    if (h == 0) {
      if (c + 1 < NCH) {  // prefetch next chunk into the other buffer
        tdm_load_v_tile(vb + (size_t)((lc0 + LCH) * N + t0) * DIMV,
                        lds_dyn_base + (((c + 1) & 1) ? OFF_V1 : OFF_V0) * 4u);
        __builtin_amdgcn_s_wait_tensorcnt(1);  // chunk c done (in-order TDM)
      } else {
        __builtin_amdgcn_s_wait_tensorcnt(0);
      }
    }
#else
    // fallback: cooperative b128 staging
    for (int idx4 = tid; idx4 < LCH * TT * DIMV / 4; idx4 += 256) {
      const int f4 = idx4 & 15;
      const int tt = (idx4 >> 4) & 7;
      const int lc = idx4 >> 7;
      ((float4*)lds_v)[idx4] =
          *(const float4*)(vb + (size_t)((lc0 + lc) * N + (t0 + tt)) * DIMV +
                           f4 * 4);
    }
#endif
    __syncthreads();  // chunk c visible to all waves

#pragma unroll
    for (int lq = 0; lq < LCH; lq += 4) {
      // A = K^T fragment: A[i=col, l'=koff(+1)], straight from global (L2-hot)
      v2f a;
      a.x = kb[(lc0 + lq + koff    ) * F + h * D + col];
      a.y = kb[(lc0 + lq + koff + 1) * F + h * D + col];
#pragma unroll
      for (int p = 0; p < NP; ++p) {
        v2f bf;
        bf.x = lds_v[((lq + koff    ) * TT + (2 * p + tsel)) * DIMV + h * DV + j];
        bf.y = lds_v[((lq + koff + 1) * TT + (2 * p + tsel)) * DIMV + h * DV + j];
        acc[p] = WMMA_F32X4(a, bf, acc[p]);
      }
    }
    __syncthreads();  // all waves done with buffer before it is re-filled
  }

  // ---- dump kv accumulators to LDS in [h][t][i][j] order ----
  {
    const int ihi = (lane >> 4) * 8;
#pragma unroll
    for (int p = 0; p < NP; ++p) {
      const int t = 2 * p + tsel;
#pragma unroll
      for (int r = 0; r < 8; ++r)
        lds_kv[(((h * TT + t) * D) + (r + ihi)) * DV + j] = acc[p][r];
    }
  }
  __syncthreads();

  // ---- Phase B: att = Q @ kv per 16-k tile; normalize; coalesced store ----
  for (int k0 = 0; k0 < N; k0 += 16) {
    v8f dacc[NP];
#pragma unroll
    for (int p = 0; p < NP; ++p)
#pragma unroll
      for (int e = 0; e < 8; ++e) dacc[p][e] = 0.f;

#pragma unroll
    for (int is = 0; is < D; is += 4) {
      v2f a;
      a.x = qb[(k0 + col) * F + h * D + is + koff];
      a.y = qb[(k0 + col) * F + h * D + is + koff + 1];
#pragma unroll
      for (int p = 0; p < NP; ++p) {
        v2f bf;
        bf.x = lds_kv[((h * TT + 2 * p + tsel) * D + is + koff    ) * DV + j];
        bf.y = lds_kv[((h * TT + 2 * p + tsel) * D + is + koff + 1) * DV + j];
        dacc[p] = WMMA_F32X4(a, bf, dacc[p]);
      }
    }

    __syncthreads();  // previous tile's copy-out done; safe to write scale/out

    // per-(h,k) normalization factor (lanes 0..15 of each wave)
    if (lane < 16) {
      float qs = 0.f;
#pragma unroll
      for (int i = 0; i < D; ++i)
        qs += qb[(k0 + lane) * F + h * D + i] * ksum[h * D + i];
      if (qs == 0.f) qs = 1e-5f;
      scale[h * 16 + lane] = 1.0f / qs;
    }

    // deposit D tile: lds_out[kk][tt][h*8+j]
    {
      const int khi = (lane >> 4) * 8;
#pragma unroll
      for (int p = 0; p < NP; ++p) {
        const int t = 2 * p + tsel;
#pragma unroll
        for (int r = 0; r < 8; ++r)
          lds_out[((r + khi) * TT + t) * DIMV + h * DV + j] = dacc[p][r];
      }
    }
    __syncthreads();

    // coalesced, normalized b128 stores: per kk a contiguous 2KB run of (t,f)
    for (int idx4 = tid; idx4 < 16 * TT * DIMV / 4; idx4 += 256) {
      const int f4 = idx4 & 15;      // float4 index within 64-float f row
      const int tt = (idx4 >> 4) & 7;
      const int kk = idx4 >> 7;
      const float s = scale[(f4 >> 1) * 16 + kk];  // h = (f4*4)>>3 = f4>>1
      float4 val = ((float4*)lds_out)[idx4];
      val.x *= s; val.y *= s; val.z *= s; val.w *= s;
      *(float4*)(ob + (size_t)((k0 + kk) * N + (t0 + tt)) * DIMV + f4 * 4) = val;
    }
  }
}

extern "C" void kernel_launch(void* const* d_in, const int* in_sizes, int n_in,
                              void* d_out, int out_size, void* d_ws, size_t ws_size,
                              hipStream_t stream) {
  (void)in_sizes; (void)n_in; (void)out_size; (void)d_ws; (void)ws_size;
  const float* q = (const float*)d_in[0];
  const float* k = (const float*)d_in[1];
  const float* v = (const float*)d_in[2];
  float* out = (float*)d_out;

  const dim3 grid(Bn * (N / TT));   // 256 blocks: (b, 8-t tile)
  const dim3 block(256);            // 8 wave32 = 8 heads
  const size_t shmem = (size_t)LDS_FLOATS * sizeof(float);  // ~129 KB dynamic LDS
  hipLaunchKernelGGL(KernelAttnCoef_17351667876294_kernel, grid, block, shmem,
                     stream, q, k, v, out);
}